// PhysNetInteraction_23905787969540
// MI455X (gfx1250) — compile-verified
//
#include <hip/hip_runtime.h>
#include <hip/hip_bf16.h>
#include <math.h>
#include <stdint.h>

// ---------------------------------------------------------------------------
// PhysNet interaction block for MI455X (gfx1250).
// fp32 matrix math via V_WMMA_F32_16X16X4_F32; staging via
// GLOBAL_LOAD_ASYNC_TO_LDS_B128 (ASYNCcnt protocol), double-buffered in LDS.
// ---------------------------------------------------------------------------

#define FDIM 128
#define NBR  64
#define NBAS 25
#define NBASP 32          // NB padded to multiple of 4 (K for WMMA)
#define ATOMS_PER_B 1024
#define NATOMS 8192       // B*A
#define NF (NATOMS * FDIM)

typedef float v2f __attribute__((ext_vector_type(2)));
typedef float v8f __attribute__((ext_vector_type(8)));

__device__ __forceinline__ float swish_f(float x) {
    return x / (1.0f + __expf(-x));
}
__device__ __forceinline__ float softplus_f(float x) {
    return (x > 20.0f) ? x : log1pf(__expf(x));
}
__device__ __forceinline__ float sign_f(float x) {
    return (float)((x > 0.0f) - (x < 0.0f));
}

// gfx1250 async global->LDS copy (bypasses VGPRs; tracked by ASYNCcnt).
// AS(3) addresses are the low 32 bits of the generic LDS pointer.
__device__ __forceinline__ void async_g2l_b128(unsigned lds_addr,
                                               unsigned long long gaddr) {
    asm volatile("global_load_async_to_lds_b128 %0, %1, off"
                 :: "v"(lds_addr), "v"(gaddr) : "memory");
}
__device__ __forceinline__ void wait_async0() {
    asm volatile("s_wait_asynccnt 0x0" ::: "memory");
}
__device__ __forceinline__ unsigned lds_off(const void* p) {
    return (unsigned)(uintptr_t)p;
}

// ---------------------------------------------------------------------------
// Y[M,F] = swish(X[M,F]) @ W[F,F] + bias[F]  (+ add[M,F] if add != nullptr)
// 256 threads (8 waves), 32 rows per block. Each wave: 16 rows x 2 col tiles.
// W is streamed in four 32-row chunks, double-buffered: chunk c+1 is
// async-DMA'd into LDS while WMMAs consume chunk c.
// Safe for in-place X == Y (block reads its rows into LDS before any store).
// ---------------------------------------------------------------------------
__global__ __launch_bounds__(256) void dense_swish_kernel(
    const float* __restrict__ X, const float* __restrict__ W,
    const float* __restrict__ bias, const float* __restrict__ add,
    float* __restrict__ Y, int M)
{
    __shared__ float ldsW[2][32 * FDIM];   // 2 x 16 KB: double-buffered W chunks
    __shared__ float ldsX[32 * FDIM];      // 16 KB: swished X tile

    const int tid     = threadIdx.x;
    const int rowBase = blockIdx.x * 32;

    // Kick off async DMA of W chunk 0 while we compute swish(X) into LDS.
    {
        unsigned long long ga = (unsigned long long)(uintptr_t)W + (unsigned)tid * 16u;
        unsigned la = lds_off(&ldsW[0][0]) + (unsigned)tid * 16u;
        #pragma unroll
        for (int it = 0; it < 4; ++it) {       // 32*128*4B / (256*16B) = 4
            async_g2l_b128(la, ga);
            la += 4096u; ga += 4096ull;
        }
    }
    for (int i = tid; i < 32 * FDIM; i += 256) {
        float xv = X[(size_t)rowBase * FDIM + i];
        ldsX[i] = swish_f(xv);
    }
    wait_async0();
    __syncthreads();

    const int wave   = tid >> 5;
    const int lane   = tid & 31;
    const int rt     = wave >> 2;            // 0..1 (row tile)
    const int ctBase = (wave & 3) * 2;       // 0,2,4,6 (col tile pair base)
    const int lrow   = rt * 16 + (lane & 15);
    const int khalf  = (lane >> 4) * 2;      // 0 or 2 (A/B K sub-select)
    const int nq     = lane & 15;

    v8f acc0 = {};
    v8f acc1 = {};

    for (int c = 0; c < 4; ++c) {            // four 32-row K chunks
        // Prefetch next chunk into the idle buffer (overlapped with WMMA).
        if (c < 3) {
            unsigned long long ga = (unsigned long long)(uintptr_t)W
                                  + (size_t)(c + 1) * 32 * FDIM * 4
                                  + (unsigned)tid * 16u;
            unsigned la = lds_off(&ldsW[(c + 1) & 1][0]) + (unsigned)tid * 16u;
            #pragma unroll
            for (int it = 0; it < 4; ++it) {
                async_g2l_b128(la, ga);
                la += 4096u; ga += 4096ull;
            }
        }
        const float* wbuf = &ldsW[c & 1][0];
        const int kc = c * 32;
        #pragma unroll
        for (int kk = 0; kk < 32; kk += 4) {
            // A fragment: 16x4 f32 (contiguous pair -> ds_load_b64)
            v2f a = *(const v2f*)&ldsX[lrow * FDIM + kc + kk + khalf];
            const int n0 = ctBase * 16 + nq;
            const int n1 = n0 + 16;
            v2f b0, b1;
            b0.x = wbuf[(kk + khalf + 0) * FDIM + n0];
            b0.y = wbuf[(kk + khalf + 1) * FDIM + n0];
            b1.x = wbuf[(kk + khalf + 0) * FDIM + n1];
            b1.y = wbuf[(kk + khalf + 1) * FDIM + n1];
            acc0 = __builtin_amdgcn_wmma_f32_16x16x4_f32(
                false, a, false, b0, (short)0, acc0, false, false);
            acc1 = __builtin_amdgcn_wmma_f32_16x16x4_f32(
                false, a, false, b1, (short)0, acc1, false, false);
        }
        wait_async0();      // next chunk landed in LDS
        __syncthreads();    // all waves done reading current buffer
    }

    // Store: C/D layout -> VGPR r: lanes0-15 M=r, lanes16-31 M=r+8; N=lane&15
    const int rofs = 8 * (lane >> 4);
    #pragma unroll
    for (int r = 0; r < 8; ++r) {
        const int row = rowBase + rt * 16 + r + rofs;
        const int c0  = ctBase * 16 + nq;
        const int c1  = c0 + 16;
        float y0 = acc0[r] + bias[c0];
        float y1 = acc1[r] + bias[c1];
        if (add) {
            y0 += add[(size_t)row * FDIM + c0];
            y1 += add[(size_t)row * FDIM + c1];
        }
        Y[(size_t)row * FDIM + c0] = y0;
        Y[(size_t)row * FDIM + c1] = y1;
    }
}

// ---------------------------------------------------------------------------
// Fused continuous-filter convolution. One workgroup per atom (8192 blocks):
//   Wf[64,128] = f_ij[64,25] @ Wfilt[25,128]   (WMMA, K padded to 32)
//   Wf *= mollifier_cutoff(r) * neighbor_mask
//   out[f] = x_i[f] + sum_n Wf[n,f] * x_j[neighbors[n], f]
// Never materializes Wf in HBM (saves 256 MB of traffic). f_ij / Wfilt are
// async-DMA'd straight into LDS; the raw f_ij staging area overlays ldsO
// (last raw read is before the pre-WMMA barrier, first ldsO write after it).
// ---------------------------------------------------------------------------
__global__ __launch_bounds__(256) void filter_conv_kernel(
    const float* __restrict__ f_ij, const float* __restrict__ Wfilt,
    const float* __restrict__ r_ij, const float* __restrict__ nmask,
    const int*   __restrict__ neighbors,
    const float* __restrict__ x_j, const float* __restrict__ x_i,
    float* __restrict__ out)
{
    __shared__ float ldsF[NBR * NBASP];     //  8 KB: f_ij tile, K padded
    __shared__ float ldsW[NBASP * FDIM];    // 16 KB: Wfilt, K padded
    __shared__ float ldsO[NBR * FDIM];      // 32 KB: filter tile (+ raw overlay)
    __shared__ float ldsCut[NBR];
    __shared__ int   ldsN[NBR];
    __shared__ float ldsRed[FDIM];

    float* ldsFraw = ldsO;                  // overlay: first 1600 floats

    const int tid  = threadIdx.x;
    const int atom = blockIdx.x;            // 0..8191
    const int b    = atom >> 10;

    // Async DMA: raw f_ij tile (64*25 contiguous floats = 400 x b128)
    {
        const unsigned long long gbase =
            (unsigned long long)(uintptr_t)(f_ij + (size_t)atom * NBR * NBAS);
        const unsigned lbase = lds_off(ldsFraw);
        for (int i = tid; i < (NBR * NBAS) / 4; i += 256)
            async_g2l_b128(lbase + (unsigned)i * 16u, gbase + (unsigned long long)i * 16ull);
    }
    // Async DMA: Wfilt rows 0..24 (25*128 contiguous floats = 800 x b128)
    {
        const unsigned long long gbase = (unsigned long long)(uintptr_t)Wfilt;
        const unsigned lbase = lds_off(ldsW);
        for (int i = tid; i < (NBAS * FDIM) / 4; i += 256)
            async_g2l_b128(lbase + (unsigned)i * 16u, gbase + (unsigned long long)i * 16ull);
    }
    // Zero-pad Wfilt rows 25..31 (disjoint from async-written region).
    for (int i = tid; i < (NBASP - NBAS) * FDIM; i += 256)
        ldsW[NBAS * FDIM + i] = 0.0f;
    // Cutoff * mask and neighbor indices
    if (tid < NBR) {
        const float r = r_ij[(size_t)atom * NBR + tid];
        float cut = 0.0f;
        if (r + 1e-7f < 5.0f) {
            const float rc = r * 0.2f;
            cut = __expf(1.0f - 1.0f / (1.0f - rc * rc));
        }
        ldsCut[tid] = cut * nmask[(size_t)atom * NBR + tid];
        ldsN[tid]   = neighbors[(size_t)atom * NBR + tid];
    }
    wait_async0();
    __syncthreads();

    // Repack f_ij with K padding 25 -> 32 (stride needed for aligned A frags).
    for (int i = tid; i < NBR * NBASP; i += 256) {
        const int n = i >> 5, k = i & 31;
        ldsF[i] = (k < NBAS) ? ldsFraw[n * NBAS + k] : 0.0f;
    }
    __syncthreads();

    // WMMA: 4 row tiles x 8 col tiles; 8 waves -> 4 accum tiles each
    const int wave  = tid >> 5;
    const int lane  = tid & 31;
    const int rt    = wave >> 1;            // 0..3
    const int cg    = (wave & 1) * 4;       // col tile group base
    const int lrow  = rt * 16 + (lane & 15);
    const int khalf = (lane >> 4) * 2;
    const int nq    = lane & 15;

    v8f acc[4] = {{}, {}, {}, {}};
    #pragma unroll
    for (int kk = 0; kk < NBASP; kk += 4) {
        v2f a = *(const v2f*)&ldsF[lrow * NBASP + kk + khalf];
        #pragma unroll
        for (int t = 0; t < 4; ++t) {
            const int n0 = (cg + t) * 16 + nq;
            v2f bfrag;
            bfrag.x = ldsW[(kk + khalf + 0) * FDIM + n0];
            bfrag.y = ldsW[(kk + khalf + 1) * FDIM + n0];
            acc[t] = __builtin_amdgcn_wmma_f32_16x16x4_f32(
                false, a, false, bfrag, (short)0, acc[t], false, false);
        }
    }

    // Scatter filter tile (scaled by cutoff) into LDS
    {
        const int rofs = 8 * (lane >> 4);
        #pragma unroll
        for (int t = 0; t < 4; ++t)
            #pragma unroll
            for (int r = 0; r < 8; ++r) {
                const int n = rt * 16 + r + rofs;      // neighbor row
                const int f = (cg + t) * 16 + nq;      // feature col
                ldsO[n * FDIM + f] = acc[t][r] * ldsCut[n];
            }
    }
    __syncthreads();

    // Gather-reduce: conv[f] = sum_n ldsO[n][f] * x_j[nbr[n]][f]
    const int f    = tid & 127;
    const int half = tid >> 7;
    float s = 0.0f;
    #pragma unroll 4
    for (int n = half * 32; n < half * 32 + 32; ++n) {
        const size_t row = (size_t)b * ATOMS_PER_B + ldsN[n];
        s += ldsO[n * FDIM + f] * x_j[row * FDIM + f];
    }
    if (half == 1) ldsRed[f] = s;
    __syncthreads();
    if (half == 0) {
        const float tot = s + ldsRed[f];
        out[(size_t)atom * FDIM + f] = x_i[(size_t)atom * FDIM + f] + tot;
    }
}

// ---------------------------------------------------------------------------
// Attention stats: per batch, w = softplus(sign(q) * <x, key>) * atom_mask;
// store per-atom weights + per-batch sums (normalization done in addfeat).
// ---------------------------------------------------------------------------
__global__ __launch_bounds__(256) void attn_stats_kernel(
    const float* __restrict__ x,
    const float* __restrict__ ckey, const float* __restrict__ skey,
    const float* __restrict__ charges, const float* __restrict__ spins,
    const float* __restrict__ amask,
    float* __restrict__ qw, float* __restrict__ sw,
    float* __restrict__ qsum, float* __restrict__ ssum)
{
    __shared__ float redq[256];
    __shared__ float reds[256];
    const int b   = blockIdx.x;
    const int tid = threadIdx.x;
    const float sgq = sign_f(charges[b]);
    const float sgs = sign_f(spins[b]);

    float lq = 0.0f, ls = 0.0f;
    #pragma unroll
    for (int a0 = 0; a0 < 4; ++a0) {
        const int a = tid * 4 + a0;
        const float* xa = x + ((size_t)b * ATOMS_PER_B + a) * FDIM;
        float dq = 0.0f, ds = 0.0f;
        for (int f = 0; f < FDIM; ++f) {
            const float xv = xa[f];
            dq += xv * ckey[f];
            ds += xv * skey[f];
        }
        const float m  = amask[b * ATOMS_PER_B + a];
        const float wq = softplus_f(sgq * dq) * m;
        const float ws = softplus_f(sgs * ds) * m;
        qw[b * ATOMS_PER_B + a] = wq;
        sw[b * ATOMS_PER_B + a] = ws;
        lq += wq; ls += ws;
    }
    redq[tid] = lq; reds[tid] = ls;
    __syncthreads();
    for (int s2 = 128; s2 > 0; s2 >>= 1) {
        if (tid < s2) { redq[tid] += redq[tid + s2]; reds[tid] += reds[tid + s2]; }
        __syncthreads();
    }
    if (tid == 0) { qsum[b] = redq[0]; ssum[b] = reds[0]; }
}

// x += q*qw_norm*cemb + s*sw_norm*semb  (in-place safe)
__global__ __launch_bounds__(256) void addfeat_kernel(
    const float* __restrict__ x2,
    const float* __restrict__ qw, const float* __restrict__ sw,
    const float* __restrict__ qsum, const float* __restrict__ ssum,
    const float* __restrict__ charges, const float* __restrict__ spins,
    const float* __restrict__ cemb, const float* __restrict__ semb,
    float* __restrict__ x3)
{
    const int i = blockIdx.x * 256 + threadIdx.x;   // 0 .. NF-1
    const int f    = i & 127;
    const int atom = i >> 7;
    const int b    = atom >> 10;
    const float wq = charges[b] * qw[atom] / (qsum[b] + 1e-8f);
    const float ws = spins[b]   * sw[atom] / (ssum[b] + 1e-8f);
    x3[i] = x2[i] + wq * cemb[f] + ws * semb[f];
}

// ---------------------------------------------------------------------------
// Host wiring. Input indices follow setup_inputs() flattening (depth-first):
//  0:x 1:r_ij 2:neighbors 3:neighbor_mask 4:f_ij 5:charges 6:spins 7:atom_mask
//  8..11:  res_in  {W1,b1,W2,b2}
//  12..15: res_i   {W1,b1,W2,b2}
//  16:Wi 17:bi
//  18..21: res_j   {W1,b1,W2,b2}
//  22:Wj 23:bj
//  24:Wfilt
//  25..28: res_v   {W1,b1,W2,b2}
//  29:Wv 30:bv
//  31..34: res_out {W1,b1,W2,b2}
//  35:charge_embedding 36:spin_embedding 37:charge_keys 38:spin_keys
// ---------------------------------------------------------------------------
extern "C" void kernel_launch(void* const* d_in, const int* in_sizes, int n_in,
                              void* d_out, int out_size, void* d_ws, size_t ws_size,
                              hipStream_t stream)
{
    const float* x_in   = (const float*)d_in[0];
    const float* r_ij   = (const float*)d_in[1];
    const int*   nbrs   = (const int*)  d_in[2];
    const float* nmask  = (const float*)d_in[3];
    const float* f_ij   = (const float*)d_in[4];
    const float* charges= (const float*)d_in[5];
    const float* spins  = (const float*)d_in[6];
    const float* amask  = (const float*)d_in[7];

    const float* Win1 = (const float*)d_in[8],  *bin1 = (const float*)d_in[9];
    const float* Win2 = (const float*)d_in[10], *bin2 = (const float*)d_in[11];
    const float* Wi1  = (const float*)d_in[12], *bi1  = (const float*)d_in[13];
    const float* Wi2  = (const float*)d_in[14], *bi2  = (const float*)d_in[15];
    const float* Wi   = (const float*)d_in[16], *bi   = (const float*)d_in[17];
    const float* Wj1  = (const float*)d_in[18], *bj1  = (const float*)d_in[19];
    const float* Wj2  = (const float*)d_in[20], *bj2  = (const float*)d_in[21];
    const float* Wj   = (const float*)d_in[22], *bj   = (const float*)d_in[23];
    const float* Wfilt= (const float*)d_in[24];
    const float* Wv1  = (const float*)d_in[25], *bv1  = (const float*)d_in[26];
    const float* Wv2  = (const float*)d_in[27], *bv2  = (const float*)d_in[28];
    const float* Wv   = (const float*)d_in[29], *bv   = (const float*)d_in[30];
    const float* Wo1  = (const float*)d_in[31], *bo1  = (const float*)d_in[32];
    const float* Wo2  = (const float*)d_in[33], *bo2  = (const float*)d_in[34];
    const float* cemb = (const float*)d_in[35];
    const float* semb = (const float*)d_in[36];
    const float* ckey = (const float*)d_in[37];
    const float* skey = (const float*)d_in[38];

    float* ws   = (float*)d_ws;
    float* H    = ws;                   // temp hidden
    float* X1   = ws + 1 * (size_t)NF;  // x after res_in
    float* XI   = ws + 2 * (size_t)NF;  // x_i -> xc (in place)
    float* XJ   = ws + 3 * (size_t)NF;  // x_j -> rv (reused)
    float* X2   = ws + 4 * (size_t)NF;  // x after conv update / after features
    float* qw   = ws + 5 * (size_t)NF;
    float* sw   = qw + NATOMS;
    float* qsum = sw + NATOMS;
    float* ssum = qsum + 8;

    const dim3 blk(256);
    const dim3 gemmGrid(NATOMS / 32);   // 256 blocks
    float* out = (float*)d_out;

    // res_in: X1 = x + W2@swish(W1@swish(x)+b1)+b2
    dense_swish_kernel<<<gemmGrid, blk, 0, stream>>>(x_in, Win1, bin1, nullptr, H, NATOMS);
    dense_swish_kernel<<<gemmGrid, blk, 0, stream>>>(H, Win2, bin2, x_in, X1, NATOMS);
    // x_i = Wi @ swish(residual(X1, res_i)) + bi
    dense_swish_kernel<<<gemmGrid, blk, 0, stream>>>(X1, Wi1, bi1, nullptr, H, NATOMS);
    dense_swish_kernel<<<gemmGrid, blk, 0, stream>>>(H, Wi2, bi2, X1, XI, NATOMS);
    dense_swish_kernel<<<gemmGrid, blk, 0, stream>>>(XI, Wi, bi, nullptr, XI, NATOMS);
    // x_j = Wj @ swish(residual(X1, res_j)) + bj
    dense_swish_kernel<<<gemmGrid, blk, 0, stream>>>(X1, Wj1, bj1, nullptr, H, NATOMS);
    dense_swish_kernel<<<gemmGrid, blk, 0, stream>>>(H, Wj2, bj2, X1, XJ, NATOMS);
    dense_swish_kernel<<<gemmGrid, blk, 0, stream>>>(XJ, Wj, bj, nullptr, XJ, NATOMS);
    // xc = x_i + cfconv(x_j)   (fused filter GEMM + cutoff + gather-reduce)
    filter_conv_kernel<<<dim3(NATOMS), blk, 0, stream>>>(
        f_ij, Wfilt, r_ij, nmask, nbrs, XJ, XI, XI);
    // v = Wv @ swish(residual(xc, res_v)) + bv ;  X2 = X1 + v
    dense_swish_kernel<<<gemmGrid, blk, 0, stream>>>(XI, Wv1, bv1, nullptr, H, NATOMS);
    dense_swish_kernel<<<gemmGrid, blk, 0, stream>>>(H, Wv2, bv2, XI, XJ, NATOMS);
    dense_swish_kernel<<<gemmGrid, blk, 0, stream>>>(XJ, Wv, bv, X1, X2, NATOMS);
    // charge / spin attention features (in-place on X2)
    attn_stats_kernel<<<dim3(8), blk, 0, stream>>>(
        X2, ckey, skey, charges, spins, amask, qw, sw, qsum, ssum);
    addfeat_kernel<<<dim3(NF / 256), blk, 0, stream>>>(
        X2, qw, sw, qsum, ssum, charges, spins, cemb, semb, X2);
    // res_out -> d_out
    dense_swish_kernel<<<gemmGrid, blk, 0, stream>>>(X2, Wo1, bo1, nullptr, H, NATOMS);
    dense_swish_kernel<<<gemmGrid, blk, 0, stream>>>(H, Wo2, bo2, X2, out, NATOMS);
}